// GlobalAssigner_53163105190159
// MI455X (gfx1250) — compile-verified
//
#include <hip/hip_runtime.h>
#include <hip/hip_bf16.h>
#include <math.h>

// Problem constants from the reference
#define BB 16
#define LL 28
#define RR 64
#define EE 34
#define DD 512

typedef __attribute__((ext_vector_type(2))) float v2f;
typedef __attribute__((ext_vector_type(8))) float v8f;

// ---------------------------------------------------------------------------
// Kernel 1: A[b,r,d] = sum_l logits[b,l,r] * arg[b,l,d]   (blocks 0..1023)
//           G[b,e]   = 0                                   (blocks 1024..1087)
// ---------------------------------------------------------------------------
__global__ __launch_bounds__(128) void prep_kernel(const float* __restrict__ logits,
                                                   const float* __restrict__ arg,
                                                   float* __restrict__ Abuf,
                                                   float* __restrict__ G) {
    int bid = blockIdx.x;
    int tid = threadIdx.x;
    if (bid < BB * RR) {
        int b = bid >> 6;      // / RR
        int r = bid & (RR - 1);
        float acc0 = 0.f, acc1 = 0.f, acc2 = 0.f, acc3 = 0.f;
        const float* lg = logits + (size_t)(b * LL) * RR + r;
        const float* ar = arg + (size_t)(b * LL) * DD;
        #pragma unroll 4
        for (int l = 0; l < LL; ++l) {
            float w = lg[l * RR];
            const float* a = ar + l * DD;
            acc0 += w * a[tid];
            acc1 += w * a[tid + 128];
            acc2 += w * a[tid + 256];
            acc3 += w * a[tid + 384];
        }
        float* o = Abuf + (size_t)(b * RR + r) * DD;
        o[tid]       = acc0;
        o[tid + 128] = acc1;
        o[tid + 256] = acc2;
        o[tid + 384] = acc3;
    } else {
        int idx = (bid - BB * RR) * 128 + tid;   // 64 blocks * 128 = 8192 = 16*512
        G[idx] = 0.0f;
    }
}

// ---------------------------------------------------------------------------
// Kernel 2: G[b,e] += sum_d evt_emb[b,d] * WTT[evt_type[b], d, e]
// grid: 16 b * 16 chunks of 32 columns, 1 wave each
// ---------------------------------------------------------------------------
__global__ __launch_bounds__(32) void evt_kernel(const float* __restrict__ evt_emb,
                                                 const int* __restrict__ evt_type,
                                                 const float* __restrict__ WTT,
                                                 float* __restrict__ G) {
    int b = blockIdx.x >> 4;
    int e = (blockIdx.x & 15) * 32 + threadIdx.x;
    int et = evt_type[b];
    const float* w = WTT + (size_t)et * DD * DD + e;
    const float* emb = evt_emb + (size_t)b * DD;
    float acc = 0.f;
    #pragma unroll 8
    for (int d = 0; d < DD; ++d) {
        acc += emb[d] * w[(size_t)d * DD];
    }
    unsafeAtomicAdd(&G[b * DD + e], acc);
}

// ---------------------------------------------------------------------------
// Kernel 3: G[b,e] += sum_{r,d} A[b,r,d] * WRT[r,d,e]  via V_WMMA_F32_16X16X4_F32
// grid: 64 r * 16 n-chunks (32 columns per wave), one wave32 per block.
// M = b (16), K = d (512 per r), N = e.
//
// Fragment layouts (CDNA5 ISA 7.12.2, 32-bit A 16x4 / 32-bit C 16x16):
//   A: lane = m + 16*(k>>1), vgpr j = k&1      -> per-lane contiguous pair in d
//   B: lane = n + 16*(k>>1), vgpr j = k&1      -> lanes coalesced over n
//   C: lanes 0-15: M=v, N=lane ; lanes 16-31: M=v+8, N=lane-16
// ---------------------------------------------------------------------------
__global__ __launch_bounds__(32) void wmma_div_kernel(const float* __restrict__ Abuf,
                                                      const float* __restrict__ WRT,
                                                      float* __restrict__ G) {
    int r    = blockIdx.x >> 4;
    int n0   = (blockIdx.x & 15) * 32;
    int lane = threadIdx.x;
    int m    = lane & 15;
    int half = lane >> 4;

    // A rows for this lane's M index, offset by the lane-half's K start
    const float* Arow = Abuf + (size_t)(m * RR + r) * DD + 2 * half;
    // B base: column n0+m of WRT[r,:,:]
    const float* Wb = WRT + (size_t)r * DD * DD + n0 + m;

    v8f c0 = {};
    v8f c1 = {};

    #pragma unroll 4
    for (int d0 = 0; d0 < DD; d0 += 4) {
        v2f a = *(const v2f*)(Arow + d0);              // A[m, d0+2*half .. +1], 8B load

        const float* wrow = Wb + (size_t)(d0 + 2 * half) * DD;
        v2f b0, b1;
        b0.x = wrow[0];            // WRT[r, d0+2h  , n0+m]
        b0.y = wrow[DD];           // WRT[r, d0+2h+1, n0+m]
        b1.x = wrow[16];           // second 16-column tile
        b1.y = wrow[DD + 16];

        c0 = __builtin_amdgcn_wmma_f32_16x16x4_f32(false, a, false, b0,
                                                   (short)0, c0, false, false);
        c1 = __builtin_amdgcn_wmma_f32_16x16x4_f32(false, a, false, b1,
                                                   (short)0, c1, false, false);
    }

    // Scatter-accumulate C into G with native f32 atomics (reduction over r blocks)
    #pragma unroll
    for (int v = 0; v < 8; ++v) {
        int row = v + 8 * half;                        // M (= b)
        unsafeAtomicAdd(&G[row * DD + n0 + m],      c0[v]);
        unsafeAtomicAdd(&G[row * DD + n0 + 16 + m], c1[v]);
    }
}

// ---------------------------------------------------------------------------
// Kernel 4: score = sigmoid( relu( (G/2) @ w1 + b1 ) @ w2 + b2 )
// One block of 1024 threads: thread t -> (b = t>>6, j = t&63) hidden unit.
// ---------------------------------------------------------------------------
__global__ __launch_bounds__(1024) void mlp_kernel(const float* __restrict__ G,
                                                   const float* __restrict__ w1,
                                                   const float* __restrict__ b1,
                                                   const float* __restrict__ w2,
                                                   const float* __restrict__ b2,
                                                   float* __restrict__ out) {
    __shared__ float hs[BB * 64];
    int t = threadIdx.x;
    int b = t >> 6;
    int j = t & 63;
    float acc = b1[j];
    const float* g = G + (size_t)b * DD;
    #pragma unroll 8
    for (int d = 0; d < DD; ++d) {
        acc += (0.5f * g[d]) * w1[d * 64 + j];
    }
    hs[t] = fmaxf(acc, 0.f);
    __syncthreads();
    if (t < BB) {
        float s = b2[0];
        #pragma unroll
        for (int k = 0; k < 64; ++k) s += hs[t * 64 + k] * w2[k];
        out[t] = 1.0f / (1.0f + expf(-s));
    }
}

// ---------------------------------------------------------------------------
extern "C" void kernel_launch(void* const* d_in, const int* in_sizes, int n_in,
                              void* d_out, int out_size, void* d_ws, size_t ws_size,
                              hipStream_t stream) {
    const float* logits   = (const float*)d_in[0];   // [B,L,R]
    const float* evt_emb  = (const float*)d_in[1];   // [B,1,D]
    const float* arg      = (const float*)d_in[2];   // [B,L,D]
    // d_in[3] arg_padding_num: multiply-then-divide by same scalar cancels -> unused
    const int*   evt_type = (const int*)d_in[4];     // [B]
    const float* WRT      = (const float*)d_in[5];   // [R,D,D]
    const float* WTT      = (const float*)d_in[6];   // [E,D,D]
    const float* w1       = (const float*)d_in[7];   // [D,64]
    const float* b1       = (const float*)d_in[8];   // [64]
    const float* w2       = (const float*)d_in[9];   // [64,1]
    const float* b2       = (const float*)d_in[10];  // [1]
    float* out = (float*)d_out;                      // [B,1]

    float* Abuf = (float*)d_ws;                      // [B,R,D]  = 2 MB
    float* G    = Abuf + (size_t)BB * RR * DD;       // [B,D]    = 32 KB

    // 1) A = einsum('blr,bld->brd') and zero the accumulator G
    prep_kernel<<<BB * RR + 64, 128, 0, stream>>>(logits, arg, Abuf, G);
    // 2) G += evt_trans
    evt_kernel<<<BB * 16, 32, 0, stream>>>(evt_emb, evt_type, WTT, G);
    // 3) G += div  (single-pass GEMM over all of WRT, fp32 WMMA)
    wmma_div_kernel<<<RR * 16, 32, 0, stream>>>(Abuf, WRT, G);
    // 4) scorer MLP
    mlp_kernel<<<1, 1024, 0, stream>>>(G, w1, b1, w2, b2, out);
}